// StrokeAlphaRenderer_6442450944615
// MI455X (gfx1250) — compile-verified
//
#include <hip/hip_runtime.h>
#include <hip/hip_bf16.h>
#include <stdint.h>

#define FONTNUM        100
#define TOPK_STROKE    5
#define N_CHAR_CLASSES 94
#define N_CHARS        4096
#define TILE_ELEMS     4096      // 64*64
#define TILE_F4        1024      // float4 per tile

// ---------------------------------------------------------------------------
// Phase 1a: per text-box row: argmax over 100 fonts (wave32 reduction, first-
// occurrence tie-break like jnp.argmax) + softmax over 5 stroke params with
// 1/255 folded into the stored weights.
// ---------------------------------------------------------------------------
__global__ void k_prep_font(const float* __restrict__ fp,   // [4096,100]
                            const float* __restrict__ sp,   // [4096,5]
                            int* __restrict__ fonts,        // [4096]
                            float* __restrict__ wmix)       // [4096,5]
{
    const int gid  = blockIdx.x * blockDim.x + threadIdx.x;
    const int row  = gid >> 5;
    const int lane = gid & 31;
    if (row >= N_CHARS) return;

    const float* r = fp + (size_t)row * FONTNUM;
    float best = -3.4e38f;
    int   bi   = 0;
    for (int j = lane; j < FONTNUM; j += 32) {
        float v = r[j];
        if (v > best) { best = v; bi = j; }
    }
    #pragma unroll
    for (int off = 16; off > 0; off >>= 1) {
        float ov = __shfl_down(best, off, 32);
        int   oi = __shfl_down(bi,   off, 32);
        if (ov > best || (ov == best && oi < bi)) { best = ov; bi = oi; }
    }
    if (lane == 0) {
        fonts[row] = bi;
        const float* s = sp + (size_t)row * TOPK_STROKE;
        float x[TOPK_STROKE];
        float m = -3.4e38f;
        #pragma unroll
        for (int k = 0; k < TOPK_STROKE; ++k) { x[k] = s[k]; m = fmaxf(m, x[k]); }
        float e[TOPK_STROKE], sum = 0.f;
        #pragma unroll
        for (int k = 0; k < TOPK_STROKE; ++k) { e[k] = expf(x[k] - m); sum += e[k]; }
        const float inv = 1.0f / (sum * 255.0f);   // fold alpha/255 normalization
        #pragma unroll
        for (int k = 0; k < TOPK_STROKE; ++k) wmix[row * TOPK_STROKE + k] = e[k] * inv;
    }
}

// ---------------------------------------------------------------------------
// Phase 1b: per char row: argmax over 94 classes (wave32 reduction).
// ---------------------------------------------------------------------------
__global__ void k_prep_char(const float* __restrict__ crv,  // [4096,94]
                            int* __restrict__ chars)        // [4096]
{
    const int gid  = blockIdx.x * blockDim.x + threadIdx.x;
    const int row  = gid >> 5;
    const int lane = gid & 31;
    if (row >= N_CHARS) return;

    const float* r = crv + (size_t)row * N_CHAR_CLASSES;
    float best = -3.4e38f;
    int   bi   = 0;
    for (int j = lane; j < N_CHAR_CLASSES; j += 32) {
        float v = r[j];
        if (v > best) { best = v; bi = j; }
    }
    #pragma unroll
    for (int off = 16; off > 0; off >>= 1) {
        float ov = __shfl_down(best, off, 32);
        int   oi = __shfl_down(bi,   off, 32);
        if (ov > best || (ov == best && oi < bi)) { best = ov; bi = oi; }
    }
    if (lane == 0) chars[row] = bi;
}

// ---------------------------------------------------------------------------
// Phase 2: one block (256 threads, 8 wave32) per char.
// Stage the five gathered 16 KB tiles into LDS with the CDNA5 async
// global->LDS data mover (ASYNCcnt path), then blend with FMAs.
// 80 KB LDS per block -> 4 blocks per WGP (320 KB), enough to cover latency.
// ---------------------------------------------------------------------------
__global__ void __launch_bounds__(256)
k_render(const float* __restrict__ table,     // [100,5,94,64,64]
         const int*   __restrict__ text_idx,  // [4096]
         const int*   __restrict__ fonts,     // [4096] per text-box
         const int*   __restrict__ chars,     // [4096] per char
         const float* __restrict__ wmix,      // [4096,5] per text-box (pre-scaled by 1/255)
         float*       __restrict__ out)       // [4096,64,64]
{
    __shared__ float4       tiles[TOPK_STROKE * TILE_F4];   // 80 KB
    __shared__ float        s_w[TOPK_STROKE];
    __shared__ const float* s_base[TOPK_STROKE];

    const int n   = blockIdx.x;
    const int tid = threadIdx.x;

    if (tid == 0) {
        const int ti = text_idx[n];
        const int f  = fonts[ti];
        const int ch = chars[n];
        const long long cb = (long long)ch * TILE_ELEMS;
        const long long fb = (long long)f * ((long long)TOPK_STROKE * N_CHAR_CLASSES * TILE_ELEMS);
        #pragma unroll
        for (int s = 0; s < TOPK_STROKE; ++s) {
            s_base[s] = table + fb + (long long)s * ((long long)N_CHAR_CLASSES * TILE_ELEMS) + cb;
            s_w[s]    = wmix[ti * TOPK_STROKE + s];
        }
    }
    __syncthreads();

    // Issue 5 tiles x 4 chunks of async b128 copies (16 B per lane per issue;
    // one wave moves 512 B per instruction). GV addressing: 64-bit vaddr.
    #pragma unroll
    for (int s = 0; s < TOPK_STROKE; ++s) {
        const char*    gb   = (const char*)s_base[s];
        const unsigned ldsb = (unsigned)(uintptr_t)(&tiles[s * TILE_F4]);
        #pragma unroll
        for (int c = 0; c < 4; ++c) {
            const unsigned off = (unsigned)(c * 256 + tid) * 16u;
            const char*    g   = gb + off;
            const unsigned l   = ldsb + off;
            asm volatile("global_load_async_to_lds_b128 %0, %1, off"
                         :: "v"(l), "v"(g) : "memory");
        }
    }
    // Each wave drains its own ASYNCcnt, then the workgroup barrier makes all
    // LDS writes visible to every wave.
    asm volatile("s_wait_asynccnt 0x0" ::: "memory");
    __syncthreads();

    const float w0 = s_w[0], w1 = s_w[1], w2 = s_w[2], w3 = s_w[3], w4 = s_w[4];
    float4* out4 = (float4*)out + (size_t)n * TILE_F4;

    #pragma unroll
    for (int c = 0; c < 4; ++c) {
        const int i4 = c * 256 + tid;
        const float4 a0 = tiles[0 * TILE_F4 + i4];
        const float4 a1 = tiles[1 * TILE_F4 + i4];
        const float4 a2 = tiles[2 * TILE_F4 + i4];
        const float4 a3 = tiles[3 * TILE_F4 + i4];
        const float4 a4 = tiles[4 * TILE_F4 + i4];
        float4 r;
        r.x = w0*a0.x + w1*a1.x + w2*a2.x + w3*a3.x + w4*a4.x;
        r.y = w0*a0.y + w1*a1.y + w2*a2.y + w3*a3.y + w4*a4.y;
        r.z = w0*a0.z + w1*a1.z + w2*a2.z + w3*a3.z + w4*a4.z;
        r.w = w0*a0.w + w1*a1.w + w2*a2.w + w3*a3.w + w4*a4.w;
        out4[i4] = r;
    }
}

// ---------------------------------------------------------------------------
// Launch glue.
// Inputs (setup_inputs order):
//   0 font_pred  f32 [8,512,100]
//   1 stroke_param_pred f32 [8,512,5]
//   2 char_rec_vec f32 [4096,94]
//   3 table f32 [100,5,94,64,64]
//   4 char_labels int (UNUSED by reference)
//   5 text_indexes int [1,4096]
// Output: f32 [4096,64,64]
// ---------------------------------------------------------------------------
extern "C" void kernel_launch(void* const* d_in, const int* in_sizes, int n_in,
                              void* d_out, int out_size, void* d_ws, size_t ws_size,
                              hipStream_t stream) {
    (void)in_sizes; (void)n_in; (void)out_size; (void)ws_size;

    const float* font_pred = (const float*)d_in[0];
    const float* stroke_pp = (const float*)d_in[1];
    const float* char_rec  = (const float*)d_in[2];
    const float* table     = (const float*)d_in[3];
    const int*   text_idx  = (const int*)d_in[5];   // char_labels (d_in[4]) unused

    float* out = (float*)d_out;

    // Workspace layout: fonts[4096] | chars[4096] | wmix[4096*5]  = 112 KB
    int*   fonts = (int*)d_ws;
    int*   chars = (int*)((char*)d_ws + 16384);
    float* wmix  = (float*)((char*)d_ws + 32768);

    // Phase 1: one wave32 per row -> 4096 waves -> 512 blocks of 256.
    k_prep_font<<<512, 256, 0, stream>>>(font_pred, stroke_pp, fonts, wmix);
    k_prep_char<<<512, 256, 0, stream>>>(char_rec, chars);

    // Phase 2: one block per char.
    k_render<<<N_CHARS, 256, 0, stream>>>(table, text_idx, fonts, chars, wmix, out);
}